// FrameEKFCPR_16535624089793
// MI455X (gfx1250) — compile-verified
//
#include <hip/hip_runtime.h>
#include <hip/hip_bf16.h>
#include <stdint.h>

// FrameEKFCPR on MI455X (gfx1250).
//
// The reference is a 16384-step sequential Kalman scan (non-associative:
// argmin slicing + atan2), so the kernel is a pure latency chain. Design:
//   * 2 blocks (one per polarization dim), each a SINGLE wave32.
//   * 8 samples per lane => the xor-butterfly reduction leaves the frame sums
//     in every lane, and every lane redundantly computes the 2x2 Kalman
//     update in registers: ZERO barriers, zero LDS state, zero broadcasts.
//   * CDNA5 async global->LDS loads double-buffer the next frames' samples
//     (16 x global_load_async_to_lds_b32 per frame, ASYNCcnt-tracked);
//     buffer readiness/reuse needs only per-wave s_wait_asynccnt/s_wait_dscnt
//     since the single wave reads exactly what it loaded.
//   * Phases within a lane are an arithmetic sequence, so only 2 sincos per
//     frame are computed; the other 7 rotations come from a 4-FMA complex
//     rotation recurrence.
//   * Output == the in-loop de-rotated samples, stored fire-and-forget.
//   * No WMMA on purpose: all matrices are 2x2 (below 16x16 granularity) and
//     QAM16 slicing on a square grid is a 4-op per-axis quantization, which
//     beats any 16-point distance matmul.

#define CPR_FRAMES 16384
#define CPR_N      256
#define CPR_DIMS   2
#define CPR_ALPHA  0.98f
#define CPR_HALF_SQRT10 1.5811388300841898f   // sqrt(10)/2
#define CPR_INV_SQRT10  0.31622776601683794f  // 1/sqrt(10)

// Low 32 bits of a generic pointer into the LDS aperture are the wave-relative
// LDS byte address (ISA 10.2).
__device__ __forceinline__ uint32_t lds_byte_addr(const void* p) {
  return (uint32_t)(uintptr_t)p;
}

// CDNA5 async copy: 4 bytes per lane, global -> LDS, tracked by ASYNCcnt.
__device__ __forceinline__ void async_g2l_b32(const float* gbase,
                                              uint32_t byte_off,
                                              uint32_t lds_addr) {
#if defined(__HIP_DEVICE_COMPILE__) && defined(__gfx1250__)
  asm volatile("global_load_async_to_lds_b32 %0, %1, %2"
               :
               : "v"(lds_addr), "v"(byte_off),
                 "s"((unsigned long long)(uintptr_t)gbase)
               : "memory");
#else
  (void)gbase; (void)byte_off; (void)lds_addr;
#endif
}

__device__ __forceinline__ void wait_asynccnt_le16() {
#if defined(__HIP_DEVICE_COMPILE__) && defined(__gfx1250__)
  asm volatile("s_wait_asynccnt 0x10" ::: "memory");
#endif
}

__device__ __forceinline__ void wait_asynccnt_0() {
#if defined(__HIP_DEVICE_COMPILE__) && defined(__gfx1250__)
  asm volatile("s_wait_asynccnt 0x0" ::: "memory");
#endif
}

__device__ __forceinline__ void wait_dscnt_0() {
#if defined(__HIP_DEVICE_COMPILE__) && defined(__gfx1250__)
  asm volatile("s_wait_dscnt 0x0" ::: "memory");
#endif
}

__device__ __forceinline__ float wave32_sum(float v) {
#pragma unroll
  for (int i = 16; i > 0; i >>= 1) v += __shfl_xor(v, i, 32);
  return v;  // full sum present in every lane (butterfly)
}

__global__ __launch_bounds__(32)
void cpr_scan_kernel(const float* __restrict__ sig_re,
                     const float* __restrict__ sig_im,
                     float* __restrict__ out /* interleaved complex64 */) {
  // Frame double-buffers; sample n of the owned dim lives at byte 4n.
  __shared__ __align__(16) float bufR[2][CPR_N];
  __shared__ __align__(16) float bufI[2][CPR_N];

  const int dpol = blockIdx.x;   // polarization dim this block owns
  const int lane = threadIdx.x;  // 0..31

  // Global byte offset of (frame f, sample n=j*32+lane, dim dpol):
  //   f*2048 + j*256 + lane*8 + dpol*4
  const uint32_t g_lane = (uint32_t)(lane * 8 + dpol * 4);

  const uint32_t ldsR0 = lds_byte_addr(&bufR[0][0]) + (uint32_t)lane * 4u;
  const uint32_t ldsR1 = lds_byte_addr(&bufR[1][0]) + (uint32_t)lane * 4u;
  const uint32_t ldsI0 = lds_byte_addr(&bufI[0][0]) + (uint32_t)lane * 4u;
  const uint32_t ldsI1 = lds_byte_addr(&bufI[1][0]) + (uint32_t)lane * 4u;

  // Issue one frame's 16 async b32 loads (8 per array); async instr j places
  // sample n=j*32+lane at LDS byte 4n, so lane L's 8 samples are contiguous.
  auto issue_frame = [&](int f, int b) {
    const uint32_t gbase = (uint32_t)f * 2048u + g_lane;
    const uint32_t lR = b ? ldsR1 : ldsR0;
    const uint32_t lI = b ? ldsI1 : ldsI0;
#pragma unroll
    for (int j = 0; j < 8; ++j) {
      const uint32_t go = gbase + (uint32_t)j * 256u;  // shared by re & im
      async_g2l_b32(sig_re, go, lR + (uint32_t)j * 128u);
      async_g2l_b32(sig_im, go, lI + (uint32_t)j * 128u);
    }
  };

  // Kalman state, replicated (identically) in every lane's registers.
  float z0 = 0.0f, z1 = 0.0f;                        // z = [phase, freq]
  float p00 = 0.f, p01 = 0.f, p10 = 0.f, p11 = 0.f;  // P
  float q00 = 0.f, q01 = 0.f, q10 = 0.f, q11 = 1e-9f;// Q0

  issue_frame(0, 0);
  issue_frame(1, 1);

  // Frame-invariant per-lane tables: centered index and its square.
  const float n_base = (float)(lane * 8) - 127.5f;
  float ncs[8], nc2[8];
#pragma unroll
  for (int j = 0; j < 8; ++j) {
    ncs[j] = n_base + (float)j;
    nc2[j] = ncs[j] * ncs[j];
  }

  // Per-lane output base: complex index k = f*512 + 16*lane + 2*j + dpol.
  float2* outp = (float2*)out + (size_t)(lane * 16 + dpol);

  for (int f = 0; f < CPR_FRAMES; ++f) {
    const int b = f & 1;

    // Oldest 16 async loads (frame f) complete for this wave.
    wait_asynccnt_le16();

    // Pull this lane's 8 samples: two b128 LDS reads per array.
    float yr[8], yi[8];
    {
      const float4* pr = (const float4*)&bufR[b][lane * 8];
      const float4* pi = (const float4*)&bufI[b][lane * 8];
      *(float4*)&yr[0] = pr[0]; *(float4*)&yr[4] = pr[1];
      *(float4*)&yi[0] = pi[0]; *(float4*)&yi[4] = pi[1];
    }
    // Reads landed in VGPRs -> safe to re-target this buffer (same wave only).
    wait_dscnt_0();
    if (f + 2 < CPR_FRAMES) issue_frame(f + 2, b);

    const float zp0 = __fmaf_rn(256.0f, z1, z0);       // z_p[0]

    // Only two sincos per frame: base phase of this lane and the step z1;
    // the remaining 7 rotations are a complex-multiply recurrence.
    const float phi0 = __fmaf_rn(n_base, z1, zp0);
    float sj, cj, sz, cz;
    __sincosf(phi0, &sj, &cj);
    __sincosf(z1, &sz, &cz);

    float S_re = 0.f, S_im = 0.f, S_ni = 0.f, S_nn = 0.f;
#pragma unroll
    for (int j = 0; j < 8; ++j) {
      const float spr = yr[j] * cj + yi[j] * sj;       // s_p = y * exp(-i phi)
      const float spi = yi[j] * cj - yr[j] * sj;

      // Output is exactly s_p; fire-and-forget b64 store (imm offset 2j).
      outp[2 * j] = make_float2(spr, spi);

      // QAM16 slice: square grid => independent per-axis quantization.
      float lr = __fmaf_rn(2.0f, floorf(spr * CPR_HALF_SQRT10), 1.0f);
      float li = __fmaf_rn(2.0f, floorf(spi * CPR_HALF_SQRT10), 1.0f);
      lr = fminf(fmaxf(lr, -3.0f), 3.0f);
      li = fminf(fmaxf(li, -3.0f), 3.0f);
      const float dr = lr * CPR_INV_SQRT10;
      const float di = li * CPR_INV_SQRT10;

      // scd = s_p * conj(d)
      const float re = spr * dr + spi * di;
      const float im = spi * dr - spr * di;

      S_re += re;
      S_im += im;
      S_ni = __fmaf_rn(ncs[j], im, S_ni);
      S_nn = __fmaf_rn(nc2[j], sqrtf(re * re + im * im), S_nn);

      // Advance the rotator: (c,s) <- (c,s) * (cos z1, sin z1).
      if (j < 7) {
        const float cn = cj * cz - sj * sz;
        const float sn = sj * cz + cj * sz;
        cj = cn; sj = sn;
      }
    }
    outp += 512;  // next frame (one pointer add per iteration)

    // Butterfly: all lanes end with the full frame sums.
    S_re = wave32_sum(S_re);
    S_im = wave32_sum(S_im);
    S_ni = wave32_sum(S_ni);
    S_nn = wave32_sum(S_nn);

    // ---- 2x2 Kalman update, computed redundantly by every lane ----------
    const float e0 = atan2f(S_im, S_re);
    const float e1 = S_ni / S_nn;

    const float Nf = 256.0f;
    const float Pp00 = p00 + Nf * (p01 + p10) + Nf * Nf * p11 + q00;
    const float Pp01 = p01 + Nf * p11 + q01;
    const float Pp10 = p10 + Nf * p11 + q10;
    const float Pp11 = p11 + q11;

    // S = P_p + R, R = diag(0.01, 1e-4); S is PD => pinv == inv
    const float S00 = Pp00 + 0.01f, S01 = Pp01;
    const float S10 = Pp10,         S11 = Pp11 + 1e-4f;
    const float idet = 1.0f / (S00 * S11 - S01 * S10);
    const float i00 =  S11 * idet, i01 = -S01 * idet;
    const float i10 = -S10 * idet, i11 =  S00 * idet;

    const float G00 = Pp00 * i00 + Pp01 * i10;
    const float G01 = Pp00 * i01 + Pp01 * i11;
    const float G10 = Pp10 * i00 + Pp11 * i10;
    const float G11 = Pp10 * i01 + Pp11 * i11;

    const float ge0 = G00 * e0 + G01 * e1;
    const float ge1 = G10 * e0 + G11 * e1;

    z0 = zp0 + ge0;                                    // z_n
    z1 = z1 + ge1;

    const float Pn00 = (1.0f - G00) * Pp00 - G01 * Pp10;
    const float Pn01 = (1.0f - G00) * Pp01 - G01 * Pp11;
    const float Pn10 = -G10 * Pp00 + (1.0f - G11) * Pp10;
    const float Pn11 = -G10 * Pp01 + (1.0f - G11) * Pp11;
    p00 = Pn00; p01 = Pn01; p10 = Pn10; p11 = Pn11;

    const float om = 1.0f - CPR_ALPHA;
    q00 = CPR_ALPHA * q00 + om * ge0 * ge0;
    q01 = CPR_ALPHA * q01 + om * ge0 * ge1;
    q10 = CPR_ALPHA * q10 + om * ge1 * ge0;
    q11 = CPR_ALPHA * q11 + om * ge1 * ge1;
  }

  wait_asynccnt_0();  // drain async queue before wave exit
}

extern "C" void kernel_launch(void* const* d_in, const int* in_sizes, int n_in,
                              void* d_out, int out_size, void* d_ws, size_t ws_size,
                              hipStream_t stream) {
  (void)in_sizes; (void)n_in; (void)out_size; (void)d_ws; (void)ws_size;
  const float* sig_re = (const float*)d_in[0];
  const float* sig_im = (const float*)d_in[1];
  // d_in[2]/d_in[3] (constellation) unused: the QAM16 square grid is sliced
  // analytically, which equals the Euclidean argmin the reference performs.
  float* out = (float*)d_out;

  // One single-wave block per polarization dim; the frame scan is inherently
  // sequential, so the entire job is two sync-free wave32 latency chains.
  cpr_scan_kernel<<<dim3(CPR_DIMS), dim3(32), 0, stream>>>(sig_re, sig_im, out);
}